// GLRFast_47433618817164
// MI455X (gfx1250) — compile-verified
//
#include <hip/hip_runtime.h>
#include <stdint.h>

// Problem constants (reference: B=2, G=8, C=4, H=W=384, 12 edges)
#define H_IMG 384
#define W_IMG 384
#define C_CH  4
#define N_E   12
#define TH    32
#define TW    32

// LDS staging regions
#define PS_H  40   // patch tile + halo 4 each side
#define PS_W  40
#define XS_H  38   // x = conv(patch): tile + halo 3
#define XS_W  38
#define XS_PW 40
#define YS_H  34   // y = x - Wx: tile + halo 1
#define YS_W  34
#define YS_PW 36
// edge weights arrive via TDM as two tiles with fixed LDS row strides:
//   A: 12 x (34-rowShift) x 32  (cols w0-1 .. w0+30, shifted at left border)
//   B: 12 x (34-rowShift) x 2   (cols w0+31, w0+32)
#define EWA_MAX (N_E * 34 * 32 + 4)
#define EWB_MAX (N_E * 34 * 2 + 4)

typedef __attribute__((ext_vector_type(4))) unsigned int u32x4;
typedef __attribute__((ext_vector_type(8))) int          i32x8;
typedef __attribute__((ext_vector_type(4))) int          i32x4;

__device__ __forceinline__ int iclamp(int v, int lo, int hi) {
  return v < lo ? lo : (v > hi ? hi : v);
}

// Low 32 bits of a generic LDS pointer == LDS byte address.
__device__ __forceinline__ uint32_t lds_addr(const void* p) {
  return (uint32_t)(uintptr_t)p;
}

// CDNA5 async global->LDS DMA (ASYNCcnt path), GV mode.
__device__ __forceinline__ void async_ld_b128(uint32_t ldsoff, const float* g) {
  asm volatile("global_load_async_to_lds_b128 %0, %1, off"
               :: "v"(ldsoff), "v"((uint64_t)(uintptr_t)g)
               : "memory");
}

// Tensor Data Mover: 3-D tile (x=cols, y=rows, z=planes) of 4-byte elements.
// D# packing per CDNA5 ISA.
// This toolchain (amdgpu-toolchain/clang-23) exposes the 6-arg builtin:
//   (u32x4 group0, i32x8 group1, i32x4 group2, i32x4 group3, i32x8 extra, i32 cpol)
__device__ __forceinline__ void tdm_load_3d(uint32_t lds_byte_off, const float* gaddr,
                                            uint32_t tensor_d0, uint32_t tensor_d1,
                                            uint32_t tensor_d2,
                                            uint32_t tile_d0, uint32_t tile_d1,
                                            uint32_t tile_d2,
                                            uint64_t stride0, uint64_t stride1) {
  uint64_t ga = (uint64_t)(uintptr_t)gaddr;
  u32x4 g0;
  g0[0] = 1u;                                              // count=1, user D#
  g0[1] = lds_byte_off;                                    // lds_addr [63:32]
  g0[2] = (uint32_t)ga;                                    // global_addr [95:64]
  g0[3] = (uint32_t)((ga >> 32) & 0x1FFFFFFu) | (2u << 30);// addr[56:32] | type=2
  i32x8 g1;
  g1[0] = (int)(2u << 16);                                 // data_size=2 (4 bytes)
  g1[1] = (int)((tensor_d0 & 0xFFFFu) << 16);              // tensor_dim0[15:0] @ [63:48]
  g1[2] = (int)((tensor_d0 >> 16) | ((tensor_d1 & 0xFFFFu) << 16));
  g1[3] = (int)((tensor_d1 >> 16) | (tile_d0 << 16));      // tile_dim0 @ [127:112]
  g1[4] = (int)(tile_d1 | (tile_d2 << 16));                // tile_dim1, tile_dim2
  g1[5] = (int)(uint32_t)(stride0 & 0xFFFFFFFFu);          // tensor_dim0_stride[31:0]
  g1[6] = (int)(uint32_t)(((stride0 >> 32) & 0xFFFFu) | ((stride1 & 0xFFFFu) << 16));
  g1[7] = (int)(uint32_t)((stride1 >> 16) & 0xFFFFFFFFu);  // tensor_dim1_stride[47:16]
  i32x4 g2;
  g2[0] = (int)tensor_d2;                                  // tensor_dim2
  g2[1] = 0;                                               // tensor_dim3
  g2[2] = 0;                                               // tensor_dim2_stride[31:0]
  g2[3] = 0;                                               // stride[47:32] | tile_dim3
  i32x4 g3 = {0, 0, 0, 0};
  i32x8 g4 = {0, 0, 0, 0, 0, 0, 0, 0};                     // unused trailing group
  __builtin_amdgcn_tensor_load_to_lds(g0, g1, g2, g3, g4, 0);
}

__global__ __launch_bounds__(256)
void glr_fused_kernel(const float* __restrict__ patchs,
                      const float* __restrict__ edgew,
                      const float* __restrict__ p01p,
                      const float* __restrict__ p02ap,
                      const float* __restrict__ p02bp,
                      const float* __restrict__ p03p,
                      float* __restrict__ out) {
  __shared__ __align__(16) float ps  [C_CH][PS_H][PS_W];  // 25.6 KB
  __shared__ __align__(16) float ewsA[EWA_MAX];           // 52.2 KB
  __shared__ __align__(16) float ewsB[EWB_MAX];           //  3.3 KB
  __shared__ __align__(16) float xs  [C_CH][XS_H][XS_PW]; // 24.3 KB
  __shared__ __align__(16) float ys  [C_CH][YS_H][YS_PW]; // 19.6 KB

  const int tid = threadIdx.x;
  const int h0  = blockIdx.y * TH;
  const int w0  = blockIdx.x * TW;
  const size_t plane = (size_t)H_IMG * W_IMG;
  const float* Pb = patchs + (size_t)blockIdx.z * C_CH * plane;
  const float* Wb = edgew  + (size_t)blockIdx.z * N_E  * plane;
  float*       Ob = out    + (size_t)blockIdx.z * C_CH * plane;

  // Border geometry for the edge-weight tiles (uniform per block).
  const int rowShift = (h0 == 0) ? 1 : 0;        // skip out-of-image top row
  const int colShift = (w0 == 0) ? 1 : 0;        // skip out-of-image left col
  const int orow  = h0 - 1 + rowShift;           // >= 0
  const int ocolA = w0 - 1 + colShift;           // >= 0
  const int ocolB = w0 + 31;                     // <= 383
  const int trows = 34 - rowShift;               // tile rows (LDS plane stride/32)

  // ---- Wave 0: launch the TDM for the dominant stream (edge weights).
  // Bottom overhang is zero-filled by the TDM via tensor_dim1, keeping the
  // LDS plane stride constant at trows*tile_dim0.
  if (tid < 32) {
    tdm_load_3d(lds_addr(&ewsA[0]), Wb + (size_t)orow * W_IMG + ocolA,
                (uint32_t)(W_IMG - ocolA), (uint32_t)(H_IMG - orow), N_E,
                32u, (uint32_t)trows, N_E,
                (uint64_t)W_IMG, (uint64_t)H_IMG * W_IMG);
    tdm_load_3d(lds_addr(&ewsB[0]), Wb + (size_t)orow * W_IMG + ocolB,
                (uint32_t)(W_IMG - ocolB), (uint32_t)(H_IMG - orow), N_E,
                2u, (uint32_t)trows, N_E,
                (uint64_t)W_IMG, (uint64_t)H_IMG * W_IMG);
  }

  // ---- All waves: async-DMA the patch tile + halo (ASYNCcnt, overlaps TDM).
  for (int t = tid; t < C_CH * PS_H * (PS_W / 4); t += 256) {  // 1600 x b128
    int q = t % (PS_W / 4);
    int r = (t / (PS_W / 4)) % PS_H;
    int c = t / ((PS_W / 4) * PS_H);
    int gr = iclamp(h0 - 4 + r,     0, H_IMG - 1);
    int gc = iclamp(w0 - 4 + 4 * q, 0, W_IMG - 4);  // 16B-aligned; clamped chunks
                                                    // only cover out-of-image cells
    async_ld_b128(lds_addr(&ps[c][r][4 * q]),
                  Pb + ((size_t)c * H_IMG + gr) * W_IMG + gc);
  }
  asm volatile("s_wait_asynccnt 0" ::: "memory");
  __syncthreads();

  // Combined 3x3 kernel: k = p01*K01 + p02a*K02A + p02b*K02B + p03*K03
  const float p01  = p01p[0];
  const float p02a = p02ap[0];
  const float p02b = p02bp[0];
  const float p03  = p03p[0];
  const float kc = p01 - p02a - p02b + 4.0f * p03;
  const float kU = -p03, kL = -p03;
  const float kR = p02a - p03, kD = p02b - p03;

  // ---- Stage 1: x = stats_conv(patch), reflect-pad-1, materialized at
  // clamp(h0-3+i) so the replicate padding of x is baked into xs.
  for (int t = tid; t < C_CH * XS_H * XS_W; t += 256) {
    int j = t % XS_W;
    int i = (t / XS_W) % XS_H;
    int c = t / (XS_W * XS_H);
    int a = iclamp(h0 - 3 + i, 0, H_IMG - 1);
    int b = iclamp(w0 - 3 + j, 0, W_IMG - 1);
    int am = (a == 0)         ? 1         : a - 1;   // reflect pad 1
    int ap = (a == H_IMG - 1) ? H_IMG - 2 : a + 1;
    int bm = (b == 0)         ? 1         : b - 1;
    int bp = (b == W_IMG - 1) ? W_IMG - 2 : b + 1;
    int ra  = a  - (h0 - 4), rb  = b  - (w0 - 4);
    int ram = am - (h0 - 4), rap = ap - (h0 - 4);
    int rbm = bm - (w0 - 4), rbp = bp - (w0 - 4);
    xs[c][i][j] = kc * ps[c][ra ][rb ]
                + kU * ps[c][ram][rb ]
                + kD * ps[c][rap][rb ]
                + kL * ps[c][ra ][rbm]
                + kR * ps[c][ra ][rbp];
  }
  if (tid < 32) __builtin_amdgcn_s_wait_tensorcnt(0);  // edge-weight TDM done
  __syncthreads();

  // ---- Stage 2: y = x - sum_e ew[e]*x(shift_e); zero outside the image
  // (implements the zero-pad of the transposed conv for free).
  constexpr int edr[N_E] = {-2,-1,-1,-1, 0, 0, 0, 0, 1, 1, 1, 2};
  constexpr int edc[N_E] = { 0,-1, 0, 1,-2,-1, 1, 2,-1, 0, 1, 0};
  const int estrA = trows * 32;   // LDS plane stride of tile A
  const int estrB = trows * 2;    // LDS plane stride of tile B
  for (int t = tid; t < C_CH * YS_H * YS_W; t += 256) {
    int j = t % YS_W;             // pw = w0-1+j
    int i = (t / YS_W) % YS_H;    // ph = h0-1+i
    int c = t / (YS_W * YS_H);
    int ph = h0 - 1 + i;
    int pw = w0 - 1 + j;
    float v = 0.0f;
    if (ph >= 0 && ph < H_IMG && pw >= 0 && pw < W_IMG) {
      int xi = i + 2, xj = j + 2;                 // xs index of (ph, pw)
      int ir = i - rowShift;                      // tile row (>=0 when in-image)
      const float* ewp;
      int estr;
      if (j < 32) { ewp = &ewsA[ir * 32 + (j - colShift)]; estr = estrA; }
      else        { ewp = &ewsB[ir * 2  + (j - 32)];       estr = estrB; }
      float acc = xs[c][xi][xj];
      #pragma unroll
      for (int e = 0; e < N_E; ++e)
        acc -= ewp[e * estr] * xs[c][xi + edr[e]][xj + edc[e]];
      v = acc;
    }
    ys[c][i][j] = v;
  }
  __syncthreads();

  // ---- Stage 3: transposed conv == correlate with spatially flipped kernel.
  const float fC = kc;
  const float fU = p02b - p03, fL = p02a - p03;
  const float fR = -p03,       fD = -p03;
  for (int t = tid; t < C_CH * TH * (TW / 4); t += 256) {
    int q = t % (TW / 4);
    int h = (t / (TW / 4)) % TH;
    int c = t / ((TW / 4) * TH);
    int i = h + 1;
    float4 v;
    #pragma unroll
    for (int u = 0; u < 4; ++u) {
      int j = 4 * q + u + 1;
      ((float*)&v)[u] = fC * ys[c][i    ][j    ]
                      + fU * ys[c][i - 1][j    ]
                      + fD * ys[c][i + 1][j    ]
                      + fL * ys[c][i    ][j - 1]
                      + fR * ys[c][i    ][j + 1];
    }
    *reinterpret_cast<float4*>(Ob + ((size_t)c * H_IMG + (h0 + h)) * W_IMG
                               + (w0 + 4 * q)) = v;
  }
}

extern "C" void kernel_launch(void* const* d_in, const int* in_sizes, int n_in,
                              void* d_out, int out_size, void* d_ws, size_t ws_size,
                              hipStream_t stream) {
  const float* patchs = (const float*)d_in[0];
  const float* edgew  = (const float*)d_in[1];
  // d_in[2] = node_degree: unused by the reference -> 113 MB of traffic saved.
  const float* p01  = (const float*)d_in[3];
  const float* p02a = (const float*)d_in[4];
  const float* p02b = (const float*)d_in[5];
  const float* p03  = (const float*)d_in[6];
  float* outp = (float*)d_out;

  const int BG = in_sizes[0] / (C_CH * H_IMG * W_IMG);  // 16
  dim3 grid(W_IMG / TW, H_IMG / TH, BG);
  glr_fused_kernel<<<grid, dim3(256, 1, 1), 0, stream>>>(
      patchs, edgew, p01, p02a, p02b, p03, outp);

  (void)n_in; (void)out_size; (void)d_ws; (void)ws_size;
}